// Model_26190710571339
// MI455X (gfx1250) — compile-verified
//
#include <hip/hip_runtime.h>
#include <math.h>

#define B_      16
#define NTILES_ 16384
#define FSZ_    2048
#define R_      5
#define TPW     8     // tiles per wave in the scores kernel

typedef __attribute__((ext_vector_type(2))) float v2f;
typedef __attribute__((ext_vector_type(8))) float v8f;

__device__ __forceinline__ float sigmoidf_fast(float x) {
  return 1.0f / (1.0f + __expf(-x));
}

__device__ __forceinline__ int imin(int a, int b) { return a < b ? a : b; }

// ---------------------------------------------------------------------------
// Kernel 1: scores[b*NTILES+n] = dot(feats[b][n][:], w_conv)   (HBM-bound)
// One wave per tile; 16 coalesced float4 loads per lane; wave shuffle reduce.
// 2 GiB streamed once -> ~92 us floor at 23.3 TB/s.
// ---------------------------------------------------------------------------
__global__ __launch_bounds__(256) void scores_kernel(
    const float* __restrict__ feats, const float* __restrict__ w,
    float* __restrict__ scores) {
  __shared__ float wl[FSZ_];
  for (int i = threadIdx.x; i < FSZ_; i += 256) wl[i] = w[i];
  __syncthreads();

  const int wid  = threadIdx.x >> 5;
  const int lane = threadIdx.x & 31;
  const float4* __restrict__ wv4 = (const float4*)wl;

  long base = ((long)blockIdx.x * 8 + wid) * TPW;
  for (int t = 0; t < TPW; ++t) {
    long tile = base + t;                         // 0 .. B*NTILES-1
    const float4* __restrict__ fp =
        (const float4*)(feats + tile * (long)FSZ_);
    float acc = 0.0f;
#pragma unroll
    for (int j = 0; j < FSZ_ / 128; ++j) {        // 16 iterations
      float4 f  = fp[j * 32 + lane];              // global_load_b128 (streamed)
      float4 wv = wv4[j * 32 + lane];             // ds_load_b128 (reused)
      acc = fmaf(f.x, wv.x, acc);
      acc = fmaf(f.y, wv.y, acc);
      acc = fmaf(f.z, wv.z, acc);
      acc = fmaf(f.w, wv.w, acc);
    }
#pragma unroll
    for (int off = 16; off > 0; off >>= 1)
      acc += __shfl_xor(acc, off, 32);
    if (lane == 0) scores[tile] = acc;
  }
}

// ---------------------------------------------------------------------------
// Kernel 2: per batch row, 5 smallest (ascending) ++ 5 largest (ascending).
// ---------------------------------------------------------------------------
__global__ __launch_bounds__(256) void select_kernel(
    const float* __restrict__ scores, float* __restrict__ minmax) {
  __shared__ float cmin[256 * R_];
  __shared__ float cmax[256 * R_];
  const int b = blockIdx.x;
  const float* __restrict__ row = scores + (long)b * NTILES_;

  float mn[R_], mx[R_];
#pragma unroll
  for (int i = 0; i < R_; ++i) { mn[i] = INFINITY; mx[i] = -INFINITY; }

  for (int idx = threadIdx.x; idx < NTILES_; idx += 256) {
    float v = row[idx];
    if (v < mn[R_ - 1]) {                      // keep 5 smallest, ascending
      mn[R_ - 1] = v;
#pragma unroll
      for (int i = R_ - 1; i > 0; --i)
        if (mn[i] < mn[i - 1]) { float t = mn[i]; mn[i] = mn[i - 1]; mn[i - 1] = t; }
    }
    if (v > mx[0]) {                           // keep 5 largest, ascending
      mx[0] = v;
#pragma unroll
      for (int i = 0; i < R_ - 1; ++i)
        if (mx[i] > mx[i + 1]) { float t = mx[i]; mx[i] = mx[i + 1]; mx[i + 1] = t; }
    }
  }
#pragma unroll
  for (int i = 0; i < R_; ++i) {
    cmin[threadIdx.x * R_ + i] = mn[i];
    cmax[threadIdx.x * R_ + i] = mx[i];
  }
  __syncthreads();

  if (threadIdx.x == 0) {
    float fmn[R_], fmx[R_];
#pragma unroll
    for (int i = 0; i < R_; ++i) { fmn[i] = INFINITY; fmx[i] = -INFINITY; }
    for (int j = 0; j < 256 * R_; ++j) {
      float v = cmin[j];
      if (v < fmn[R_ - 1]) {
        fmn[R_ - 1] = v;
#pragma unroll
        for (int i = R_ - 1; i > 0; --i)
          if (fmn[i] < fmn[i - 1]) { float t = fmn[i]; fmn[i] = fmn[i - 1]; fmn[i - 1] = t; }
      }
      v = cmax[j];
      if (v > fmx[0]) {
        fmx[0] = v;
#pragma unroll
        for (int i = 0; i < R_ - 1; ++i)
          if (fmx[i] > fmx[i + 1]) { float t = fmx[i]; fmx[i] = fmx[i + 1]; fmx[i + 1] = t; }
      }
    }
#pragma unroll
    for (int i = 0; i < R_; ++i) {
      minmax[b * 2 * R_ + i]      = fmn[i];   // vals[:, :R]
      minmax[b * 2 * R_ + R_ + i] = fmx[i];   // vals[:, -R:]
    }
  }
}

// ---------------------------------------------------------------------------
// Kernel 3: MLP via V_WMMA_F32_16X16X4_F32 (exact f32 matrix path).
//   X[16,10] -> sigmoid(X W1^T + b1)[16,200] -> sigmoid(.. W2^T + b2)[16,100]
//   -> .. W3^T + b3 -> logits[16], probs[16]
// Fragment layouts per CDNA5 ISA 7.12.2:
//   A (16x4 f32): M = lane&15; VGPRv holds K = (lane>>4)*2 + v
//   B (4x16 f32): N = lane&15; VGPRv holds K = (lane>>4)*2 + v (mirror of A)
//   C/D (16x16):  VGPRv -> M = v + 8*(lane>>4), N = lane&15
// All out-of-range fragment elements use clamped-address loads * mask so the
// compute loops are completely branch-free (EXEC stays all-1s around WMMA).
// ---------------------------------------------------------------------------
__global__ __launch_bounds__(128) void mlp_kernel(
    const float* __restrict__ X,
    const float* __restrict__ W1, const float* __restrict__ b1,
    const float* __restrict__ W2, const float* __restrict__ b2,
    const float* __restrict__ W3, const float* __restrict__ b3,
    float* __restrict__ out) {
  __shared__ float Xp[16][12];     // K padded 10 -> 12
  __shared__ float H1[16][208];    // N padded 200 -> 208
  __shared__ float H2[16][112];    // N padded 100 -> 112

  const int tid = threadIdx.x;
  for (int i = tid; i < 16 * 12; i += 128) {
    int m = i / 12, k = i % 12;
    Xp[m][k] = (k < 10) ? X[m * 10 + k] : 0.0f;
  }
  __syncthreads();

  const int lane  = tid & 31;
  const int wid   = __builtin_amdgcn_readfirstlane(tid >> 5);  // scalar wave id
  const int mrow  = lane & 15;          // A-fragment row
  const int ncol  = lane & 15;          // B/C/D column within tile
  const int khalf = (lane >> 4) << 1;   // 0 or 2
  const int mbase = (lane >> 4) << 3;   // 0 or 8 (C/D row base)

  // ---- Layer 1: [16,10] x [10,200] -> H1 ----
  for (int nt = wid; nt < 13; nt += 4) {
    const int   n    = nt * 16 + ncol;
    const int   nc   = imin(n, 199);
    const float nmsk = (n < 200) ? 1.0f : 0.0f;
    v8f c = {0.f, 0.f, 0.f, 0.f, 0.f, 0.f, 0.f, 0.f};
#pragma unroll
    for (int k0 = 0; k0 < 12; k0 += 4) {
      const int ka = k0 + khalf;
      v2f a; a.x = Xp[mrow][ka]; a.y = Xp[mrow][ka + 1];
      v2f bb;   // clamped address + multiplicative mask: no exec-masked loads
      bb.x = W1[nc * 10 + imin(ka,     9)] * ((ka     < 10) ? nmsk : 0.0f);
      bb.y = W1[nc * 10 + imin(ka + 1, 9)] * ((ka + 1 < 10) ? nmsk : 0.0f);
      c = __builtin_amdgcn_wmma_f32_16x16x4_f32(
              false, a, false, bb, (short)0, c, false, false);
    }
    const float bias = b1[nc] * nmsk;
#pragma unroll
    for (int v = 0; v < 8; ++v)
      H1[mbase + v][n] = sigmoidf_fast(c[v] + bias);
  }
  __syncthreads();

  // ---- Layer 2: [16,200] x [200,100] -> H2 ----
  for (int nt = wid; nt < 7; nt += 4) {
    const int   n    = nt * 16 + ncol;
    const int   nc   = imin(n, 99);
    const float nmsk = (n < 100) ? 1.0f : 0.0f;
    v8f c = {0.f, 0.f, 0.f, 0.f, 0.f, 0.f, 0.f, 0.f};
    for (int k0 = 0; k0 < 200; k0 += 4) {
      const int ka = k0 + khalf;
      v2f a; a.x = H1[mrow][ka]; a.y = H1[mrow][ka + 1];
      v2f bb;
      bb.x = W2[nc * 200 + ka    ] * nmsk;
      bb.y = W2[nc * 200 + ka + 1] * nmsk;
      c = __builtin_amdgcn_wmma_f32_16x16x4_f32(
              false, a, false, bb, (short)0, c, false, false);
    }
    const float bias = b2[nc] * nmsk;
#pragma unroll
    for (int v = 0; v < 8; ++v)
      H2[mbase + v][n] = sigmoidf_fast(c[v] + bias);
  }
  __syncthreads();

  // ---- Layer 3: [16,100] x [100,1] + sigmoid ----
  if (tid < 16) {
    float acc = b3[0];
    for (int k = 0; k < 100; ++k) acc = fmaf(H2[tid][k], W3[k], acc);
    out[tid]      = acc;                  // logits
    out[16 + tid] = sigmoidf_fast(acc);   // probs
  }
}

// ---------------------------------------------------------------------------
extern "C" void kernel_launch(void* const* d_in, const int* in_sizes, int n_in,
                              void* d_out, int out_size, void* d_ws, size_t ws_size,
                              hipStream_t stream) {
  const float* feats  = (const float*)d_in[0];
  const float* w_conv = (const float*)d_in[1];
  const float* W1     = (const float*)d_in[2];
  const float* b1     = (const float*)d_in[3];
  const float* W2     = (const float*)d_in[4];
  const float* b2     = (const float*)d_in[5];
  const float* W3     = (const float*)d_in[6];
  const float* b3     = (const float*)d_in[7];
  float* out = (float*)d_out;

  float* scores = (float*)d_ws;                       // B*NTILES floats (1 MiB)
  float* minmax = scores + (size_t)B_ * NTILES_;      // B*2R floats

  const int nblocks = (B_ * NTILES_) / (8 * TPW);     // 4096
  scores_kernel<<<nblocks, 256, 0, stream>>>(feats, w_conv, scores);
  select_kernel<<<B_, 256, 0, stream>>>(scores, minmax);
  mlp_kernel<<<1, 128, 0, stream>>>(minmax, W1, b1, W2, b2, W3, b3, out);
}